// BrainGNN_68959994904998
// MI455X (gfx1250) — compile-verified
//
#include <hip/hip_runtime.h>

// ---------------------------------------------------------------------------
// BrainGNN forward on MI455X: 3x GraphConv (agg@Wrel + b + h@Wroot) + 2 FC.
// fp32 end-to-end (working set < 192MB L2; L2-bandwidth bound, no downcast).
// GEMMs: V_WMMA_F32_16X16X4_F32, A-operand bands staged into LDS by the
// Tensor Data Mover (tensor_load_to_lds + s_wait_tensorcnt).
// Aggregation: CSR gather (built once/launch, ~1.3M int atomics + LDS scan)
// instead of 246M global float atomics; buckets sorted -> deterministic.
// ---------------------------------------------------------------------------

#define N_NODES 40000
#define N_EDGES 640000
#define D_FEAT  128
#define SCAN_B  256
#define NBLK    ((N_NODES + SCAN_B - 1) / SCAN_B)   // 157

typedef unsigned int       u32;
typedef unsigned long long u64;
typedef __attribute__((ext_vector_type(2))) float v2f;
typedef __attribute__((ext_vector_type(8))) float v8f;
typedef __attribute__((ext_vector_type(4))) u32  v4u;
typedef __attribute__((ext_vector_type(8))) int  v8i;
typedef __attribute__((ext_vector_type(4))) int  v4i;

// ---------------- TDM: 2-D tile load Global->LDS ----------------------------
// D# per ISA cdna5 ch.8: group0 = {count|flags, lds_addr, gaddr_lo,
// gaddr_hi|type=2}; group1 = {data_size=4B, tensor_dim0/1, tile_dim0/1,
// dim0 stride}. 2-D tensor -> remaining groups zero. 6-arg builtin
// (this toolchain): (v4u, v8i, v4i, v4i, v8i, i32 cpol).
__device__ inline void tdm_load_rows(const float* gsrc, u32 lds_addr,
                                     u32 kelems, u32 rows) {
    u64 ga = (u64)(uintptr_t)gsrc;
    v4u g0;
    g0.x = 1u;                                            // count=1, user D#
    g0.y = lds_addr;                                      // LDS byte address
    g0.z = (u32)ga;                                       // global addr [31:0]
    g0.w = (u32)((ga >> 32) & 0x01FFFFFFu) | (2u << 30);  // addr[56:32]|type=2
    v8i g1;
    g1[0] = (int)(2u << 16);         // wg_mask=0, data_size=2 (4 bytes)
    g1[1] = (int)(kelems << 16);     // tensor_dim0[15:0]   (bits 63:48)
    g1[2] = (int)(rows   << 16);     // tensor_dim1[15:0]   (bits 95:80)
    g1[3] = (int)(kelems << 16);     // tile_dim0           (bits 127:112)
    g1[4] = (int)rows;               // tile_dim1           (bits 143:128)
    g1[5] = (int)kelems;             // tensor_dim0_stride  (bits 191:160)
    g1[6] = 0;
    g1[7] = 0;
    v4i z4 = {0, 0, 0, 0};
    v8i z8 = {0, 0, 0, 0, 0, 0, 0, 0};
    __builtin_amdgcn_tensor_load_to_lds(g0, g1, z4, z4, z8, 0);
}

// ---------------- utility: zero int buffer ----------------------------------
__global__ void zero_i(int* __restrict__ p, int n) {
    int i = blockIdx.x * blockDim.x + threadIdx.x;
    if (i < n) p[i] = 0;
}

// ---------------- CSR build: degree count -----------------------------------
__global__ __launch_bounds__(256)
void count_edges(const int* __restrict__ dst, int* __restrict__ cnt) {
    int e = blockIdx.x * blockDim.x + threadIdx.x;
    if (e < N_EDGES)
        __hip_atomic_fetch_add(&cnt[dst[e]], 1, __ATOMIC_RELAXED, __HIP_MEMORY_SCOPE_AGENT);
}

// ---------------- CSR build: exclusive scan (LDS) ---------------------------
__global__ __launch_bounds__(SCAN_B)
void scan_block(const int* __restrict__ cnt, int* __restrict__ row_start,
                int* __restrict__ partial) {
    __shared__ int sm[SCAN_B];
    const int t = threadIdx.x;
    const int i = blockIdx.x * SCAN_B + t;
    int v = (i < N_NODES) ? cnt[i] : 0;
    sm[t] = v;
    __syncthreads();
    for (int off = 1; off < SCAN_B; off <<= 1) {
        int x = (t >= off) ? sm[t - off] : 0;
        __syncthreads();
        sm[t] += x;
        __syncthreads();
    }
    if (i < N_NODES) row_start[i] = sm[t] - v;
    if (t == SCAN_B - 1) partial[blockIdx.x] = sm[t];
}

__global__ __launch_bounds__(SCAN_B)
void scan_partial(int* __restrict__ partial) {
    __shared__ int sm[SCAN_B];
    const int t = threadIdx.x;
    int v = (t < NBLK) ? partial[t] : 0;
    sm[t] = v;
    __syncthreads();
    for (int off = 1; off < SCAN_B; off <<= 1) {
        int x = (t >= off) ? sm[t - off] : 0;
        __syncthreads();
        sm[t] += x;
        __syncthreads();
    }
    if (t < NBLK) partial[t] = sm[t] - v;
}

__global__ __launch_bounds__(SCAN_B)
void add_offsets(int* __restrict__ row_start, const int* __restrict__ partial) {
    int i = blockIdx.x * SCAN_B + threadIdx.x;
    if (i < N_NODES) row_start[i] += partial[blockIdx.x];
}

// ---------------- CSR build: bucket fill + deterministic sort ---------------
__global__ __launch_bounds__(256)
void fill_csr(const int* __restrict__ src, const int* __restrict__ dst,
              const int* __restrict__ row_start, int* __restrict__ cursor,
              int* __restrict__ csr_src) {
    int e = blockIdx.x * blockDim.x + threadIdx.x;
    if (e >= N_EDGES) return;
    int d = dst[e];
    int pos = __hip_atomic_fetch_add(&cursor[d], 1, __ATOMIC_RELAXED, __HIP_MEMORY_SCOPE_AGENT);
    csr_src[row_start[d] + pos] = src[e];
}

__global__ __launch_bounds__(256)
void sort_buckets(int* __restrict__ csr_src, const int* __restrict__ row_start,
                  const int* __restrict__ cnt) {
    int i = blockIdx.x * blockDim.x + threadIdx.x;
    if (i >= N_NODES) return;
    int s0 = row_start[i], d = cnt[i];
    for (int a = 1; a < d; ++a) {
        int key = csr_src[s0 + a];
        int b = a - 1;
        while (b >= 0 && csr_src[s0 + b] > key) {
            csr_src[s0 + b + 1] = csr_src[s0 + b];
            --b;
        }
        csr_src[s0 + b + 1] = key;
    }
}

// ---------------- aggregation: agg[i] = sum_{j in N(i)} h[src_j] ------------
// one wave/node; lanes cover the 128-float row as float4 chunks; 32 src ids
// staged per load, broadcast via __shfl (ds_bpermute).
__global__ __launch_bounds__(256)
void aggregate(const float* __restrict__ h, const int* __restrict__ csr_src,
               const int* __restrict__ row_start, const int* __restrict__ cnt,
               float* __restrict__ agg) {
    const int lane = threadIdx.x & 31;
    const int node = (blockIdx.x * blockDim.x + threadIdx.x) >> 5;
    if (node >= N_NODES) return;
    const int start = row_start[node];
    const int end   = start + cnt[node];

    float4 acc = make_float4(0.f, 0.f, 0.f, 0.f);
    const int c = lane * 4;
    for (int j = start; j < end; j += 32) {
        int myidx = j + lane;
        int s = (myidx < end) ? csr_src[myidx] : 0;
        int m = end - j; if (m > 32) m = 32;
        for (int t = 0; t < m; ++t) {
            int sv = __shfl(s, t, 32);
            float4 v = *(const float4*)(h + (size_t)sv * D_FEAT + c);
            acc.x += v.x; acc.y += v.y; acc.z += v.z; acc.w += v.w;
        }
    }
    *(float4*)(agg + (size_t)node * D_FEAT + c) = acc;
}

// ---------------- WMMA fp32 GEMM with TDM-staged A bands --------------------
// One block per 64-row M band; wave w handles N-tile w (blockDim = (N/16)*32).
// Wave 0 TDM-loads the A (and A2) 64xK band into LDS (32KB each), waits
// tensorcnt, barrier publishes to the block. Each wave then runs the K-loop
// from LDS fragments (ds_load_b64) with one global B-fragment load per
// K-step shared by 4 independent WMMA accumulators (64x16 tile per wave).
// EXEC all-ones at every WMMA (wave-uniform control flow only).
//
// ISA 7.12.2 fragment layouts (wave32):
//   A (16x4 f32):  lanes 0-15: M=lane, K=k0+{0,1}; lanes 16-31: K=k0+{2,3}
//   B (4x16 f32):  lanes 0-15: N=lane, K=k0+{0,1}; lanes 16-31: K=k0+{2,3}
//   C/D (16x16):   v[r]: lanes 0-15 M=r, N=lane; lanes 16-31 M=r+8
template <bool DUAL, bool RELU>
__global__ __launch_bounds__(256)
void gemm_wmma(const float* __restrict__ A,  const float* __restrict__ W,
               const float* __restrict__ A2, const float* __restrict__ W2,
               const float* __restrict__ bias, float* __restrict__ out,
               int N, int K) {
    __shared__ float smA[64 * D_FEAT];                 // 32 KB
    __shared__ float smA2[DUAL ? 64 * D_FEAT : 4];     // 32 KB when dual
    const int lane = threadIdx.x & 31;
    const int wave = threadIdx.x >> 5;
    const int half = lane >> 4;
    const int l    = lane & 15;
    const int base = blockIdx.y * 64;                  // 64-row M band

    if (wave == 0) {   // TDM ignores EXEC: issue from exactly one wave
        tdm_load_rows(A + (size_t)base * K, (u32)(uintptr_t)(void*)smA, K, 64);
        if (DUAL)
            tdm_load_rows(A2 + (size_t)base * K, (u32)(uintptr_t)(void*)smA2, K, 64);
        __builtin_amdgcn_s_wait_tensorcnt(0);
    }
    __syncthreads();

    const int col = wave * 16 + l;                     // this wave's N-tile
    v8f acc0, acc1, acc2, acc3;
    const float bv = bias[col];
#pragma unroll
    for (int r = 0; r < 8; ++r) { acc0[r] = bv; acc1[r] = bv; acc2[r] = bv; acc3[r] = bv; }

    const float* sA0 = smA  + (size_t)l * D_FEAT;      // rows l, l+16, l+32, l+48
    const float* sB0 = DUAL ? (smA2 + (size_t)l * D_FEAT) : nullptr;

    for (int k0 = 0; k0 < K; k0 += 4) {
        const int ka = k0 + half * 2;
        v2f b;
        b.x = W[(size_t)ka * N + col];
        b.y = W[(size_t)(ka + 1) * N + col];
        v2f a0 = *(const v2f*)(sA0 + ka);
        v2f a1 = *(const v2f*)(sA0 + 16 * D_FEAT + ka);
        v2f a2 = *(const v2f*)(sA0 + 32 * D_FEAT + ka);
        v2f a3 = *(const v2f*)(sA0 + 48 * D_FEAT + ka);
        acc0 = __builtin_amdgcn_wmma_f32_16x16x4_f32(false, a0, false, b, (short)0, acc0, false, false);
        acc1 = __builtin_amdgcn_wmma_f32_16x16x4_f32(false, a1, false, b, (short)0, acc1, false, false);
        acc2 = __builtin_amdgcn_wmma_f32_16x16x4_f32(false, a2, false, b, (short)0, acc2, false, false);
        acc3 = __builtin_amdgcn_wmma_f32_16x16x4_f32(false, a3, false, b, (short)0, acc3, false, false);
        if (DUAL) {
            v2f b2;
            b2.x = W2[(size_t)ka * N + col];
            b2.y = W2[(size_t)(ka + 1) * N + col];
            v2f c0 = *(const v2f*)(sB0 + ka);
            v2f c1 = *(const v2f*)(sB0 + 16 * D_FEAT + ka);
            v2f c2 = *(const v2f*)(sB0 + 32 * D_FEAT + ka);
            v2f c3 = *(const v2f*)(sB0 + 48 * D_FEAT + ka);
            acc0 = __builtin_amdgcn_wmma_f32_16x16x4_f32(false, c0, false, b2, (short)0, acc0, false, false);
            acc1 = __builtin_amdgcn_wmma_f32_16x16x4_f32(false, c1, false, b2, (short)0, acc1, false, false);
            acc2 = __builtin_amdgcn_wmma_f32_16x16x4_f32(false, c2, false, b2, (short)0, acc2, false, false);
            acc3 = __builtin_amdgcn_wmma_f32_16x16x4_f32(false, c3, false, b2, (short)0, acc3, false, false);
        }
    }

#pragma unroll
    for (int r = 0; r < 8; ++r) {
        if (RELU) {
            acc0[r] = acc0[r] > 0.f ? acc0[r] : 0.f;
            acc1[r] = acc1[r] > 0.f ? acc1[r] : 0.f;
            acc2[r] = acc2[r] > 0.f ? acc2[r] : 0.f;
            acc3[r] = acc3[r] > 0.f ? acc3[r] : 0.f;
        }
        const int row = base + half * 8 + r;
        out[(size_t)(row     ) * N + col] = acc0[r];
        out[(size_t)(row + 16) * N + col] = acc1[r];
        out[(size_t)(row + 32) * N + col] = acc2[r];
        out[(size_t)(row + 48) * N + col] = acc3[r];
    }
}

// ---------------- final FC: 64 -> 1 -----------------------------------------
__global__ __launch_bounds__(256)
void fc_out(const float* __restrict__ h, const float* __restrict__ w,
            const float* __restrict__ b, float* __restrict__ out, int K) {
    int i = blockIdx.x * blockDim.x + threadIdx.x;
    if (i >= N_NODES) return;
    float s = b[0];
    const float* hr = h + (size_t)i * K;
#pragma unroll 8
    for (int k = 0; k < K; ++k) s += hr[k] * w[k];
    out[i] = s;
}

// ---------------------------------------------------------------------------
extern "C" void kernel_launch(void* const* d_in, const int* in_sizes, int n_in,
                              void* d_out, int out_size, void* d_ws, size_t ws_size,
                              hipStream_t stream) {
    const float* x    = (const float*)d_in[0];
    const int*   ei   = (const int*)d_in[1];     // (2, N_EDGES): row0=src, row1=dst
    const int*   src  = ei;
    const int*   dst  = ei + N_EDGES;
    const float* Wrel0  = (const float*)d_in[2];
    const float* Wroot0 = (const float*)d_in[3];
    const float* b0     = (const float*)d_in[4];
    const float* Wrel1  = (const float*)d_in[5];
    const float* Wroot1 = (const float*)d_in[6];
    const float* b1     = (const float*)d_in[7];
    const float* Wrel2  = (const float*)d_in[8];
    const float* Wroot2 = (const float*)d_in[9];
    const float* b2     = (const float*)d_in[10];
    const float* Wfc0   = (const float*)d_in[11];
    const float* bfc0   = (const float*)d_in[12];
    const float* Wfc1   = (const float*)d_in[13];
    const float* bfc1   = (const float*)d_in[14];
    float* out = (float*)d_out;

    // ---- workspace layout ----
    const size_t nf = (size_t)N_NODES * D_FEAT;
    float* buf0 = (float*)d_ws;
    float* buf1 = buf0 + nf;
    float* agg  = buf1 + nf;
    int* cnt       = (int*)(agg + nf);
    int* cursor    = cnt + N_NODES;
    int* row_start = cursor + N_NODES;
    int* partial   = row_start + N_NODES;
    int* csr_src   = partial + SCAN_B;

    const int egrid = (N_EDGES + 255) / 256;
    const int ngrid = (N_NODES + 255) / 256;

    // ---- build CSR once (edges identical for all 3 conv layers) ----
    zero_i<<<(2 * N_NODES + 255) / 256, 256, 0, stream>>>(cnt, 2 * N_NODES); // cnt+cursor
    count_edges<<<egrid, 256, 0, stream>>>(dst, cnt);
    scan_block<<<NBLK, SCAN_B, 0, stream>>>(cnt, row_start, partial);
    scan_partial<<<1, SCAN_B, 0, stream>>>(partial);
    add_offsets<<<NBLK, SCAN_B, 0, stream>>>(row_start, partial);
    fill_csr<<<egrid, 256, 0, stream>>>(src, dst, row_start, cursor, csr_src);
    sort_buckets<<<ngrid, 256, 0, stream>>>(csr_src, row_start, cnt);

    const dim3 gband(1, N_NODES / 64);             // 625 M bands of 64 rows
    const int  agrid = (N_NODES * 32 + 255) / 256; // one wave per node

    // ---- conv0: h1 = relu(agg(x)@Wrel0 + b0 + x@Wroot0) -> buf0
    aggregate<<<agrid, 256, 0, stream>>>(x, csr_src, row_start, cnt, agg);
    gemm_wmma<true, true><<<gband, 256, 0, stream>>>(agg, Wrel0, x, Wroot0, b0,
                                                     buf0, D_FEAT, D_FEAT);
    // ---- conv1: h2 = relu(agg(h1)@Wrel1 + b1 + h1@Wroot1) -> buf1
    aggregate<<<agrid, 256, 0, stream>>>(buf0, csr_src, row_start, cnt, agg);
    gemm_wmma<true, true><<<gband, 256, 0, stream>>>(agg, Wrel1, buf0, Wroot1, b1,
                                                     buf1, D_FEAT, D_FEAT);
    // ---- conv2 (no relu): h3 = agg(h2)@Wrel2 + b2 + h2@Wroot2 -> buf0
    aggregate<<<agrid, 256, 0, stream>>>(buf1, csr_src, row_start, cnt, agg);
    gemm_wmma<true, false><<<gband, 256, 0, stream>>>(agg, Wrel2, buf1, Wroot2, b2,
                                                      buf0, D_FEAT, D_FEAT);
    // ---- fc0: h4 = relu(h3@Wfc0 + bfc0) -> buf1 (N=64, 4 waves/block)
    gemm_wmma<false, true><<<gband, 128, 0, stream>>>(buf0, Wfc0, nullptr, nullptr,
                                                      bfc0, buf1, 64, D_FEAT);
    // ---- fc1: out = h4@Wfc1 + bfc1 (N=1)
    fc_out<<<(N_NODES + 255) / 256, 256, 0, stream>>>(buf1, Wfc1, bfc1, out, 64);
}